// TopKRouter_42691974922246
// MI455X (gfx1250) — compile-verified
//
#include <hip/hip_runtime.h>
#include <hip/hip_bf16.h>

// ---------------------------------------------------------------------------
// TopK (k=1) MoE router for MI455X (gfx1250, wave32, WMMA).
// B=4, S=8192, D=1024, E=64, capacity=640.
// ---------------------------------------------------------------------------

typedef __attribute__((ext_vector_type(2))) float v2f;
typedef __attribute__((ext_vector_type(8))) float v8f;

#define RB 4
#define RS 8192
#define RD 1024
#define RE 64
#define RCAP 640
#define RM (RB * RS)          // 32768 rows
#define RBSE (RM * RE)        // 2097152 elements per output tensor

// ---------------------------------------------------------------------------
// Kernel 1: logits[M,64] = x[M,1024] @ W[64,1024]^T via V_WMMA_F32_16X16X4_F32.
// One wave owns a 16-row strip and all 4 column tiles (x streamed exactly once).
// ---------------------------------------------------------------------------
__global__ __launch_bounds__(256) void router_gemm_wmma(
    const float* __restrict__ x, const float* __restrict__ W,
    float* __restrict__ logits) {
  const int lane = threadIdx.x & 31;
  const int wave = threadIdx.x >> 5;
  const int tileM = blockIdx.x * 8 + wave;   // 0..2047
  const int rowBase = tileM * 16;
  const int m  = lane & 15;                  // M (for A) / N (for B) within tile
  const int hi = lane >> 4;                  // half-wave selects K pair

  // A fragment: lane holds {A[m][k+2*hi], A[m][k+2*hi+1]}
  const float* aPtr = x + (size_t)(rowBase + m) * RD + 2 * hi;
  // B fragment (B = W^T): lane holds {W[n][k+2*hi], W[n][k+2*hi+1]}
  const float* bPtr0 = W + (size_t)(0 * 16 + m) * RD + 2 * hi;
  const float* bPtr1 = W + (size_t)(1 * 16 + m) * RD + 2 * hi;
  const float* bPtr2 = W + (size_t)(2 * 16 + m) * RD + 2 * hi;
  const float* bPtr3 = W + (size_t)(3 * 16 + m) * RD + 2 * hi;

  v8f c0 = {}, c1 = {}, c2 = {}, c3 = {};

  for (int k = 0; k < RD; k += 16) {
#pragma unroll
    for (int kk = 0; kk < 16; kk += 4) {
      const int ko = k + kk;
      v2f a  = *(const v2f*)(aPtr  + ko);
      v2f b0 = *(const v2f*)(bPtr0 + ko);
      v2f b1 = *(const v2f*)(bPtr1 + ko);
      v2f b2 = *(const v2f*)(bPtr2 + ko);
      v2f b3 = *(const v2f*)(bPtr3 + ko);
      c0 = __builtin_amdgcn_wmma_f32_16x16x4_f32(false, a, false, b0, (short)0, c0, false, false);
      c1 = __builtin_amdgcn_wmma_f32_16x16x4_f32(false, a, false, b1, (short)0, c1, false, false);
      c2 = __builtin_amdgcn_wmma_f32_16x16x4_f32(false, a, false, b2, (short)0, c2, false, false);
      c3 = __builtin_amdgcn_wmma_f32_16x16x4_f32(false, a, false, b3, (short)0, c3, false, false);
    }
  }

  // C/D layout: VGPR r -> row (r + 8*hi), col = lane&15 within each N tile.
  float* out = logits + (size_t)(rowBase + 8 * hi) * RE + m;
#pragma unroll
  for (int r = 0; r < 8; ++r) {
    out[(size_t)r * RE + 0]  = c0[r];
    out[(size_t)r * RE + 16] = c1[r];
    out[(size_t)r * RE + 32] = c2[r];
    out[(size_t)r * RE + 48] = c3[r];
  }
}

// ---------------------------------------------------------------------------
// Kernel 2: per token argmax + top softmax prob (= 1 / sum(exp(l - max))).
// One wave per token, 2 logits per lane, butterfly reductions.
// ---------------------------------------------------------------------------
__global__ __launch_bounds__(256) void topk_softmax(
    const float* __restrict__ logits, int* __restrict__ topIdx,
    float* __restrict__ topVal) {
  const int lane = threadIdx.x & 31;
  const int token = blockIdx.x * 8 + (threadIdx.x >> 5);

  const float2 v = *(const float2*)(logits + (size_t)token * RE + 2 * lane);
  float mx;
  int idx;
  if (v.x >= v.y) { mx = v.x; idx = 2 * lane; }      // ties -> first index
  else            { mx = v.y; idx = 2 * lane + 1; }
#pragma unroll
  for (int off = 16; off > 0; off >>= 1) {
    float om = __shfl_xor(mx, off, 32);
    int oi = __shfl_xor(idx, off, 32);
    if (om > mx || (om == mx && oi < idx)) { mx = om; idx = oi; }
  }
  float s = __expf(v.x - mx) + __expf(v.y - mx);
#pragma unroll
  for (int off = 16; off > 0; off >>= 1) s += __shfl_xor(s, off, 32);

  if (lane == 0) {
    topIdx[token] = idx;
    topVal[token] = 1.0f / s;
  }
}

// ---------------------------------------------------------------------------
// Kernel 3: sequential capacity scan. One wave per (batch, expert):
// ballot-based running rank along S; flag tokens within capacity.
// Each token matches exactly one expert -> okFlag fully written, race-free.
// ---------------------------------------------------------------------------
__global__ __launch_bounds__(256) void capacity_scan(
    const int* __restrict__ topIdx, int* __restrict__ okFlag) {
  const int lane = threadIdx.x & 31;
  const int wg = blockIdx.x * 8 + (threadIdx.x >> 5);   // 0..255
  const int b = wg >> 6;
  const int e = wg & 63;
  const int base = b * RS;
  const unsigned ltMask = (1u << lane) - 1u;

  int running = 0;
  for (int s0 = 0; s0 < RS; s0 += 32) {
    const int s = s0 + lane;
    const bool match = (topIdx[base + s] == e);
    const unsigned mask = (unsigned)__ballot(match);    // wave32: low 32 bits
    if (match) {
      const int prio = running + __popc(mask & ltMask) + 1;
      okFlag[base + s] = (prio <= RCAP) ? 1 : 0;
    }
    running += __popc(mask);
  }
}

// ---------------------------------------------------------------------------
// Kernel 4: write expert_indices (int32 one-hot) and router_probs (f32).
// 16 threads per token, int4/float4 stores -> fully coalesced 16 MB.
// ---------------------------------------------------------------------------
__global__ __launch_bounds__(256) void scatter_outputs(
    const int* __restrict__ topIdx, const float* __restrict__ topVal,
    const int* __restrict__ okFlag, int* __restrict__ ei,
    float* __restrict__ rp) {
  const int t = blockIdx.x * 256 + threadIdx.x;   // 0..524287
  const int token = t >> 4;
  const int j = t & 15;

  const int idx = topIdx[token];
  const int ok = okFlag[token];
  const float val = topVal[token];
  const int col = j << 2;

  int4 vi;
  float4 vf;
  vi.x = (ok && idx == col + 0) ? 1 : 0;
  vi.y = (ok && idx == col + 1) ? 1 : 0;
  vi.z = (ok && idx == col + 2) ? 1 : 0;
  vi.w = (ok && idx == col + 3) ? 1 : 0;
  vf.x = vi.x ? val : 0.0f;
  vf.y = vi.y ? val : 0.0f;
  vf.z = vi.z ? val : 0.0f;
  vf.w = vi.w ? val : 0.0f;

  ((int4*)ei)[(size_t)token * 16 + j] = vi;
  ((float4*)rp)[(size_t)token * 16 + j] = vf;
}

// ---------------------------------------------------------------------------
extern "C" void kernel_launch(void* const* d_in, const int* in_sizes, int n_in,
                              void* d_out, int out_size, void* d_ws, size_t ws_size,
                              hipStream_t stream) {
  (void)in_sizes; (void)n_in; (void)out_size; (void)ws_size;

  const float* x = (const float*)d_in[0];   // [4, 8192, 1024] f32
  const float* W = (const float*)d_in[1];   // [64, 1024] f32
  // d_in[2] = top_k == 1 (unused; layout assumes k==1 like the reference math)

  // Outputs concatenated flat in return order:
  int*   ei = (int*)d_out;                          // expert_indices [B,S,E] int32
  float* rp = (float*)d_out + (size_t)RBSE;         // router_probs   [B,S,E] f32
  float* lg = (float*)d_out + 2 * (size_t)RBSE;     // logits         [B,S,E] f32

  // Workspace: idx[32768] int, val[32768] f32, ok[32768] int  (384 KB)
  int*   topIdx = (int*)d_ws;
  float* topVal = (float*)((char*)d_ws + (size_t)RM * 4);
  int*   okFlag = (int*)((char*)d_ws + (size_t)RM * 8);

  router_gemm_wmma<<<RM / (16 * 8), 256, 0, stream>>>(x, W, lg);
  topk_softmax<<<RM / 8, 256, 0, stream>>>(lg, topIdx, topVal);
  capacity_scan<<<(RB * RE) / 8, 256, 0, stream>>>(topIdx, okFlag);
  scatter_outputs<<<(RM * 16) / 256, 256, 0, stream>>>(topIdx, topVal, okFlag, ei, rp);
}